// LPCLitePytorch_82454782148723
// MI455X (gfx1250) — compile-verified
//
#include <hip/hip_runtime.h>
#include <hip/hip_bf16.h>

// ---------------- problem constants ----------------
#define BB   64
#define FF   3000          // frames
#define FSH  80            // frame shift = samples per coeff frame
#define LL   (FF * FSH)    // 240000
#define ORD  16            // a[0] == 1, taps a[1..15]
#define TAPS 15
#define EMPH 0.97f
#define CHUNK 25           // frames composed per wave (divides FF)
#define NC   (FF / CHUNK)  // 120 chunks per batch element

// ---------------- workspace layout (floats) ----------------
#define OFF_H    ((size_t)0)                          // B*F*80 impulse responses
#define OFF_S    (OFF_H   + (size_t)BB*FF*FSH)        // B*F*16 per-frame states
#define OFF_MC   (OFF_S   + (size_t)BB*FF*16)         // B*NC*256 chunk composites
#define OFF_SVC  (OFF_MC  + (size_t)BB*NC*256)        // B*NC*16 chunk-start states
#define OFF_DCAR (OFF_SVC + (size_t)BB*NC*16)         // B*F per-frame deemph tail
#define OFF_CARR (OFF_DCAR+ (size_t)BB*FF)            // B*F deemph carries

typedef float v2f __attribute__((ext_vector_type(2)));
typedef float v8f __attribute__((ext_vector_type(8)));

// ---------------- K1: per-frame impulse response, one frame per lane ----------------
__global__ void k_h(const float* __restrict__ lpc, float* __restrict__ h, int nFrames) {
    int t = blockIdx.x * blockDim.x + threadIdx.x;
    if (t >= nFrames) return;
    const float* a = lpc + (size_t)t * ORD;
    float ac[TAPS];
#pragma unroll
    for (int k = 0; k < TAPS; ++k) ac[k] = a[k + 1];
    float hist[TAPS];
#pragma unroll
    for (int k = 0; k < TAPS; ++k) hist[k] = 0.f;
    float* hp = h + (size_t)t * FSH;
    for (int i = 0; i < FSH; ++i) {
        float val = (i == 0) ? 1.f : 0.f;
#pragma unroll
        for (int k = 0; k < TAPS; ++k) val = fmaf(-ac[k], hist[k], val);  // hist[k]=h[i-1-k]
#pragma unroll
        for (int k = TAPS - 1; k > 0; --k) hist[k] = hist[k - 1];
        hist[0] = val;
        hp[i] = val;
    }
}

// ---------------- K2: per-frame zero-state response y_zs = T(h) @ e -> d_out ----------------
__global__ void k_yzs(const float* __restrict__ excit, const float* __restrict__ h,
                      float* __restrict__ yzs) {
    int fr = blockIdx.x;            // b*F + f
    __shared__ float hs[FSH], es[FSH];
    const float* hp = h + (size_t)fr * FSH;
    const float* ep = excit + (size_t)fr * FSH;
    for (int i = threadIdx.x; i < FSH; i += blockDim.x) { hs[i] = hp[i]; es[i] = ep[i]; }
    __syncthreads();
    int i = threadIdx.x;
    if (i < FSH) {
        float acc = 0.f;
        for (int j = 0; j <= i; ++j) acc = fmaf(hs[i - j], es[j], acc);
        yzs[(size_t)fr * FSH + i] = acc;
    }
}

// Build augmented 16x16 frame matrix A-hat into LDS (row-major [i*16+c]).
// rows 0..14: state update; col 15: zero-state tail v; row 15 = e15^T.
__device__ __forceinline__ void build_A(const float* hs, const float* as,
                                        const float* __restrict__ vrow,
                                        float* Ah, int lane) {
    for (int t = lane; t < 256; t += 32) {
        int i = t >> 4, c = t & 15;
        float val;
        if (i < 15) {
            if (c < 15) {
                float acc = 0.f;
                for (int j = 0; j <= c; ++j) acc = fmaf(hs[65 + i - j], as[j + 15 - c], acc);
                val = -acc;                               // M[i][c]
            } else {
                val = vrow[65 + i];                       // v[i] = y_zs[65+i]
            }
        } else {
            val = (c == 15) ? 1.f : 0.f;
        }
        Ah[t] = val;
    }
}

// ---------------- K3: chunk composition via WMMA (16x16x16 f32 = 4x V_WMMA_F32_16X16X4_F32) ----
__global__ void k_compose(const float* __restrict__ lpc, const float* __restrict__ h,
                          const float* __restrict__ yzs, float* __restrict__ Mc) {
    int blk = blockIdx.x;           // b*NC + nc
    int b = blk / NC, nc = blk % NC;
    int f0 = nc * CHUNK;
    int lane = threadIdx.x;
    __shared__ float hs[FSH];
    __shared__ float as[16];
    __shared__ float Ah[256];
    __shared__ float Pl[256];
    const int mrow = (lane & 16) ? 8 : 0;   // C/D layout: VGPR r -> row r (+8 upper half-wave)
    const int ncol = lane & 15;

    v8f P;
#pragma unroll
    for (int r = 0; r < 8; ++r) P[r] = ((r + mrow) == ncol) ? 1.f : 0.f;  // identity

    for (int ff = 0; ff < CHUNK; ++ff) {
        size_t fr = (size_t)b * FF + (f0 + ff);
        for (int i = lane; i < FSH; i += 32) hs[i] = h[fr * FSH + i];
        if (lane < 16) as[lane] = lpc[fr * ORD + lane];
        __syncthreads();
        build_A(hs, as, yzs + fr * FSH, Ah, lane);
#pragma unroll
        for (int r = 0; r < 8; ++r) Pl[(r + mrow) * 16 + ncol] = P[r];
        __syncthreads();

        v8f C;
#pragma unroll
        for (int r = 0; r < 8; ++r) C[r] = 0.f;
#pragma unroll
        for (int s = 0; s < 4; ++s) {
            int kb = 4 * s + ((lane & 16) ? 2 : 0);     // A 16x4 layout: v0=K0/K2, v1=K1/K3
            v2f A, Bm;
            A[0]  = Ah[(lane & 15) * 16 + kb];
            A[1]  = Ah[(lane & 15) * 16 + kb + 1];
            Bm[0] = Pl[kb * 16 + ncol];                 // B 4x16: rows striped across lanes
            Bm[1] = Pl[(kb + 1) * 16 + ncol];
            C = __builtin_amdgcn_wmma_f32_16x16x4_f32(
                    false, A, false, Bm, (short)0, C, false, false);
        }
        P = C;                                          // P <- A_hat * P
        __syncthreads();
    }
    float* mout = Mc + (size_t)blk * 256;
#pragma unroll
    for (int r = 0; r < 8; ++r) mout[(r + mrow) * 16 + ncol] = P[r];
}

// ---------------- K4: serial scan of chunk composites per batch element ----------------
__global__ void k_chunkscan(const float* __restrict__ Mc, float* __restrict__ svc) {
    int b = blockIdx.x, lane = threadIdx.x;
    __shared__ float sv[16], nv[16];
    if (lane < 16) sv[lane] = (lane == 15) ? 1.f : 0.f;   // zero state, homogeneous 1
    __syncthreads();
    for (int nc = 0; nc < NC; ++nc) {
        if (lane < 16) svc[((size_t)b * NC + nc) * 16 + lane] = sv[lane];
        const float* M = Mc + ((size_t)b * NC + nc) * 256;
        if (lane < 16) {
            float acc = 0.f;
            for (int c = 0; c < 16; ++c) acc = fmaf(M[lane * 16 + c], sv[c], acc);
            nv[lane] = acc;
        }
        __syncthreads();
        if (lane < 16) sv[lane] = nv[lane];
        __syncthreads();
    }
}

// ---------------- K5: per-frame states inside each chunk ----------------
__global__ void k_framestates(const float* __restrict__ lpc, const float* __restrict__ h,
                              const float* __restrict__ yzs, const float* __restrict__ svc,
                              float* __restrict__ sst) {
    int blk = blockIdx.x;
    int b = blk / NC, nc = blk % NC;
    int lane = threadIdx.x;
    __shared__ float hs[FSH];
    __shared__ float as[16];
    __shared__ float sv[16], nv[16];
    if (lane < 16) sv[lane] = svc[(size_t)blk * 16 + lane];
    for (int ff = 0; ff < CHUNK; ++ff) {
        size_t fr = (size_t)b * FF + (nc * CHUNK + ff);
        for (int i = lane; i < FSH; i += 32) hs[i] = h[fr * FSH + i];
        if (lane < 16) as[lane] = lpc[fr * ORD + lane];
        __syncthreads();
        if (lane < 16) sst[fr * 16 + lane] = sv[lane];    // state at frame START
        if (lane < 15) {
            float acc = yzs[fr * FSH + 65 + lane];        // v term (sv[15]==1)
            for (int c = 0; c < 15; ++c) {
                float g = 0.f;
                for (int j = 0; j <= c; ++j) g = fmaf(hs[65 + lane - j], as[j + 15 - c], g);
                acc = fmaf(-g, sv[c], acc);
            }
            nv[lane] = acc;
        }
        __syncthreads();
        if (lane < 15) sv[lane] = nv[lane];               // sv[15] stays 1
        __syncthreads();
    }
}

// ---------------- K6: outputs y, de-emphasis zero-state w (in-place in d_out) ----------------
__global__ void k_out(const float* __restrict__ lpc, const float* __restrict__ h,
                      const float* __restrict__ sst, float* __restrict__ io,
                      float* __restrict__ dcar) {
    int fr = blockIdx.x, t = threadIdx.x;
    __shared__ float hs[FSH];
    __shared__ float as[16];
    __shared__ float sv[16];
    __shared__ float yb[FSH];
    __shared__ float pw[FSH];
    for (int i = t; i < FSH; i += blockDim.x) hs[i] = h[(size_t)fr * FSH + i];
    if (t < 16) { as[t] = lpc[(size_t)fr * ORD + t]; sv[t] = sst[(size_t)fr * 16 + t]; }
    if (t == 0) { float p = 1.f; for (int d = 0; d < FSH; ++d) { pw[d] = p; p *= EMPH; } }
    __syncthreads();
    if (t < FSH) {
        float acc = io[(size_t)fr * FSH + t];             // y_zs
        for (int c = 0; c < 15; ++c) {
            float g = 0.f;
            int jmax = (c < t) ? c : t;
            for (int j = 0; j <= jmax; ++j) g = fmaf(hs[t - j], as[j + 15 - c], g);
            acc = fmaf(-g, sv[c], acc);
        }
        yb[t] = acc;
    }
    __syncthreads();
    if (t < FSH) {
        float w = 0.f;
        for (int j = 0; j <= t; ++j) w = fmaf(pw[t - j], yb[j], w);
        io[(size_t)fr * FSH + t] = w;
        if (t == FSH - 1) dcar[fr] = w;
    }
}

// ---------------- K7: de-emphasis carry scan (wave32 affine shuffle scan) ----------------
__global__ void k_carr(const float* __restrict__ dcar, float* __restrict__ carr) {
    int b = blockIdx.x, lane = threadIdx.x;
    float Q = 1.f;
    for (int d = 0; d < FSH; ++d) Q *= EMPH;              // EMPH^80
    float Qlane = 1.f;
    for (int d = 0; d < lane; ++d) Qlane *= Q;            // Q^lane
    float Q32 = 1.f;
    for (int d = 0; d < 32; ++d) Q32 *= Q;                // Q^32 (~0, stable)
    float c0 = 0.f;
    for (int f0 = 0; f0 < FF; f0 += 32) {
        int f = f0 + lane;
        float dl = (f < FF) ? dcar[(size_t)b * FF + f] : 0.f;
        float A = Q, Bv = dl;                             // map z -> A*z + B
#pragma unroll
        for (int d = 1; d < 32; d <<= 1) {
            float Ap = __shfl_up(A, d, 32);
            float Bp = __shfl_up(Bv, d, 32);
            if (lane >= d) { Bv = fmaf(A, Bp, Bv); A = A * Ap; }
        }
        float Sprev = __shfl_up(Bv, 1, 32);               // S_{lane-1}
        if (lane == 0) Sprev = 0.f;
        float cl = fmaf(Qlane, c0, Sprev);                // carry at frame start
        if (f < FF) carr[(size_t)b * FF + f] = cl;
        float S31 = __shfl(Bv, 31, 32);
        c0 = fmaf(Q32, c0, S31);
    }
}

// ---------------- K8: finalize z[n] = w[n] + EMPH^{i+1} * carry ----------------
__global__ void k_final(const float* __restrict__ carr, float* __restrict__ io, int total) {
    int n = blockIdx.x * blockDim.x + threadIdx.x;
    if (n >= total) return;
    int fr = n / FSH, i = n % FSH;
    float p = __powf(EMPH, (float)(i + 1));
    io[n] = fmaf(p, carr[fr], io[n]);
}

extern "C" void kernel_launch(void* const* d_in, const int* in_sizes, int n_in,
                              void* d_out, int out_size, void* d_ws, size_t ws_size,
                              hipStream_t stream) {
    (void)in_sizes; (void)n_in; (void)out_size; (void)ws_size;
    const float* excit = (const float*)d_in[0];           // (B, L, 1) f32
    const float* lpc   = (const float*)d_in[1];           // (B, F, 16) f32
    float* out = (float*)d_out;                           // (B, L, 1) f32
    float* ws  = (float*)d_ws;

    float* h    = ws + OFF_H;
    float* sst  = ws + OFF_S;
    float* Mc   = ws + OFF_MC;
    float* svc  = ws + OFF_SVC;
    float* dcar = ws + OFF_DCAR;
    float* carr = ws + OFF_CARR;

    const int nFrames = BB * FF;
    k_h<<<(nFrames + 255) / 256, 256, 0, stream>>>(lpc, h, nFrames);
    k_yzs<<<nFrames, 96, 0, stream>>>(excit, h, out);       // d_out used as y_zs scratch
    k_compose<<<BB * NC, 32, 0, stream>>>(lpc, h, out, Mc); // WMMA affine composition
    k_chunkscan<<<BB, 32, 0, stream>>>(Mc, svc);
    k_framestates<<<BB * NC, 32, 0, stream>>>(lpc, h, out, svc, sst);
    k_out<<<nFrames, 96, 0, stream>>>(lpc, h, sst, out, dcar);
    k_carr<<<BB, 32, 0, stream>>>(dcar, carr);
    k_final<<<(BB * LL + 255) / 256, 256, 0, stream>>>(carr, out, BB * LL);
}